// TokenEmbedding_7739531067774
// MI455X (gfx1250) — compile-verified
//
#include <hip/hip_runtime.h>
#include <hip/hip_bf16.h>
#include <stdint.h>

#ifndef DIM
#define DIM 128          // num_dim
#endif

#define TPW 8            // tokens per wave per block-iteration
#define WAVES 8          // waves per block (256 threads, wave32)

// clang-native vector types (required by __builtin_nontemporal_*)
typedef float __attribute__((ext_vector_type(4))) f32x4;
typedef int   __attribute__((ext_vector_type(4))) i32x4;

// ---------------------------------------------------------------------------
// Kernel 1: fused transpose + bias.  Wt[v][d] = W[d][v] + b[d]
// W: [DIM, vocab] row-major.  Wt: [vocab, DIM] row-major (512B rows).
// 32x32 LDS tile transpose so reads (along v) and writes (along d) coalesce.
// Wt stays with default (RT) policy: it is the L2-resident reuse set.
// ---------------------------------------------------------------------------
__global__ void build_table_kernel(const float* __restrict__ W,
                                   const float* __restrict__ b,
                                   float* __restrict__ Wt,
                                   int vocab) {
    __shared__ float tile[32][33];           // +1 pad: bank-conflict free
    const int vBase = blockIdx.x * 32;
    const int dBase = blockIdx.y * 32;
    const int x = threadIdx.x;               // 0..31

    #pragma unroll
    for (int yy = threadIdx.y; yy < 32; yy += 8) {
        const int v = vBase + x;
        const int d = dBase + yy;
        tile[yy][x] = (v < vocab && d < DIM)
                        ? W[(size_t)d * vocab + v] : 0.0f;
    }
    __syncthreads();

    const int d = dBase + x;
    const float bias = (d < DIM) ? b[d] : 0.0f;
    #pragma unroll
    for (int yy = threadIdx.y; yy < 32; yy += 8) {
        const int v = vBase + yy;
        if (v < vocab && d < DIM)
            Wt[(size_t)v * DIM + d] = tile[x][yy] + bias;
    }
}

// ---------------------------------------------------------------------------
// Kernel 2: async gather.  One wave32 per token-row; lane l moves bytes
// [16l, 16l+16) of the 512B row.  TPW rows in flight per wave via
// global_load_async_to_lds_b128 (ASYNCcnt), drained after s_wait_asynccnt 0.
// Output is written with non-temporal stores (write-once stream) so the
// L2-resident Wt table is not evicted by the 128 MiB output stream.
// Block = 256 threads = 8 waves; LDS staging = 8 waves * 8 slots * 512B = 32KB.
// ---------------------------------------------------------------------------
__global__ void gather_async_kernel(const int* __restrict__ tokens,
                                    const float* __restrict__ Wt,
                                    float* __restrict__ out,
                                    int n_tokens) {
    __shared__ __align__(16) float buf[WAVES * TPW * DIM];   // 32 KB
    const int lane = threadIdx.x & 31;
    const int wave = threadIdx.x >> 5;
    const int base = blockIdx.x * (WAVES * TPW) + wave * TPW;

    // low 32 bits of a flat shared-aperture address are the LDS offset
    const uint32_t ldsBase =
        (uint32_t)(uintptr_t)(&buf[0]) + (uint32_t)(wave * TPW * DIM * 4);
    const uint32_t laneByte = (uint32_t)(lane * 16);

    if (base + TPW <= n_tokens) {
        // token ids: streamed once -> non-temporal load
        const i32x4 ta = __builtin_nontemporal_load(
            reinterpret_cast<const i32x4*>(tokens + base));
        const i32x4 tb = __builtin_nontemporal_load(
            reinterpret_cast<const i32x4*>(tokens + base + 4));
        int t[TPW] = { ta.x, ta.y, ta.z, ta.w, tb.x, tb.y, tb.z, tb.w };

        // TPW async 512B row-fills in flight per wave (ASYNCcnt-tracked)
        #pragma unroll
        for (int k = 0; k < TPW; ++k) {
            const float*  src = Wt + (size_t)t[k] * DIM + lane * 4;
            const uint32_t ld = ldsBase + (uint32_t)(k * DIM * 4) + laneByte;
            asm volatile("global_load_async_to_lds_b128 %0, %1, off"
                         :: "v"(ld), "v"(src) : "memory");
        }

        asm volatile("s_wait_asynccnt 0x0" ::: "memory");

        const int slotBase = wave * TPW * DIM + lane * 4;
        #pragma unroll
        for (int k = 0; k < TPW; ++k) {
            const f32x4 v = *reinterpret_cast<const f32x4*>(
                &buf[slotBase + k * DIM]);
            // write-once output stream -> non-temporal store (keep Wt in L2)
            __builtin_nontemporal_store(
                v, reinterpret_cast<f32x4*>(
                       out + (size_t)(base + k) * DIM + lane * 4));
        }
    } else {
        // tail: direct copy, no async
        for (int k = 0; k < TPW; ++k) {
            const int row = base + k;
            if (row < n_tokens) {
                const int tk = tokens[row];
                const f32x4 v = *reinterpret_cast<const f32x4*>(
                    Wt + (size_t)tk * DIM + lane * 4);
                __builtin_nontemporal_store(
                    v, reinterpret_cast<f32x4*>(
                           out + (size_t)row * DIM + lane * 4));
            }
        }
    }
}

// ---------------------------------------------------------------------------
// Fallback (workspace too small): direct strided gather from W.
// ---------------------------------------------------------------------------
__global__ void gather_direct_kernel(const int* __restrict__ tokens,
                                     const float* __restrict__ W,
                                     const float* __restrict__ b,
                                     float* __restrict__ out,
                                     int n_tokens, int vocab) {
    const size_t idx = (size_t)blockIdx.x * blockDim.x + threadIdx.x;
    const size_t total = (size_t)n_tokens * DIM;
    if (idx >= total) return;
    const int d = (int)(idx & (DIM - 1));
    const int row = (int)(idx >> 7);
    const int t = tokens[row];
    out[idx] = W[(size_t)d * vocab + t] + b[d];
}

extern "C" void kernel_launch(void* const* d_in, const int* in_sizes, int n_in,
                              void* d_out, int out_size, void* d_ws, size_t ws_size,
                              hipStream_t stream) {
    const int*   tokens = (const int*)d_in[0];
    const float* W      = (const float*)d_in[1];
    const float* b      = (const float*)d_in[2];
    float*       out    = (float*)d_out;

    const int n_tokens = in_sizes[0];             // B*S = 262144
    const int vocab    = in_sizes[1] / DIM;       // 32000
    const size_t tbl_bytes = (size_t)vocab * DIM * sizeof(float);

    if (ws_size >= tbl_bytes) {
        float* Wt = (float*)d_ws;

        dim3 tgrid((vocab + 31) / 32, (DIM + 31) / 32);
        dim3 tblk(32, 8);
        build_table_kernel<<<tgrid, tblk, 0, stream>>>(W, b, Wt, vocab);

        const int tokensPerBlock = WAVES * TPW;   // 64
        const int blocks = (n_tokens + tokensPerBlock - 1) / tokensPerBlock;
        gather_async_kernel<<<blocks, 256, 0, stream>>>(tokens, Wt, out, n_tokens);
    } else {
        const size_t total = (size_t)n_tokens * DIM;
        const int blocks = (int)((total + 255) / 256);
        gather_direct_kernel<<<blocks, 256, 0, stream>>>(tokens, W, b, out,
                                                         n_tokens, vocab);
    }
}